// HierarchicalSDEParticleFilter_40613210751383
// MI455X (gfx1250) — compile-verified
//
#include <hip/hip_runtime.h>
#include <math.h>

#define KSTEPS 400
#define UPART  8192
#define DLAT   16
#define MOBS_  32
#define DT_F   0.01f
#define NBLK   32
#define NTHR   256
#define NWAVE  (NTHR/32)

typedef __attribute__((ext_vector_type(2))) float v2f;
typedef __attribute__((ext_vector_type(8))) float v8f;

__device__ __forceinline__ void ld16(float* dst, const float* __restrict__ src) {
  const float4* s = (const float4*)src;
#pragma unroll
  for (int q = 0; q < 4; ++q) {
    float4 t = s[q];
    dst[4*q+0] = t.x; dst[4*q+1] = t.y; dst[4*q+2] = t.z; dst[4*q+3] = t.w;
  }
}

__device__ __forceinline__ void st16(float* __restrict__ dst, const float* src) {
  float4* d = (float4*)dst;
#pragma unroll
  for (int q = 0; q < 4; ++q)
    d[q] = make_float4(src[4*q+0], src[4*q+1], src[4*q+2], src[4*q+3]);
}

// Monotone-counter grid barrier: all 32 blocks are co-resident on MI455X.
// Release: every thread fences its stores to L2; Acquire: tid0 spins with an
// agent-scope acquire load (invalidates the WGP-shared L0), then block barrier.
__device__ __forceinline__ void grid_sync(unsigned int* bar, unsigned int* ep_sh) {
  __threadfence();
  __syncthreads();
  if (threadIdx.x == 0) {
    unsigned int target = (*ep_sh + 1u) * (unsigned int)NBLK;
    atomicAdd(bar, 1u);
    while (__hip_atomic_load(bar, __ATOMIC_ACQUIRE, __HIP_MEMORY_SCOPE_AGENT) < target) {
      __builtin_amdgcn_s_sleep(1);
    }
    *ep_sh = *ep_sh + 1u;
  }
  __syncthreads();
}

__launch_bounds__(NTHR, 1)
__global__ void pf_persistent(
    const float* __restrict__ Z,      const float* __restrict__ x0,
    const float* __restrict__ y0,     const float* __restrict__ tau0,
    const float* __restrict__ m0,     const float* __restrict__ tau_pool,
    const float* __restrict__ m_pool, const float* __restrict__ eps_x,
    const float* __restrict__ eps_y,  const float* __restrict__ resample_u,
    const float* __restrict__ Wmat,   const float* __restrict__ logR,
    const float* __restrict__ lsx,    const float* __restrict__ lsy,
    float* __restrict__ out, unsigned int* bar, float* wsf)
{
  // -------- workspace layout (floats, after 256B header) --------
  float* gmax = wsf;                      // [32] per-block max(lw)
  float* s1a  = wsf + 32;                 // [32] per-block sum(wt)
  float* s2a  = wsf + 64;                 // [32] per-block sum(wt^2)
  float* cum  = wsf + 128;                // [8192] normalized cdf
  float* xs   = cum + UPART;              // [8192*16] resample staging
  float* ys   = xs + UPART*DLAT;
  float* ms   = ys + UPART*DLAT;
  float* tls  = ms + UPART*DLAT;          // [8192]
  float* trs  = tls + UPART;              // [8192]

  const int tid  = threadIdx.x;
  const int blk  = blockIdx.x;
  const int u    = blk * NTHR + tid;      // one particle per thread
  const int lane = tid & 31;
  const int wid  = tid >> 5;
  const int ln   = lane & 15;
  const int h    = lane >> 4;

  __shared__ __align__(16) float yv[NWAVE][32*DLAT];  // staged y  [particle][dim]
  __shared__ __align__(16) float yw[NWAVE][32*DLAT];  // staged w*y
  __shared__ float lwl[NWAVE][32];
  __shared__ float mom[DLAT + DLAT*DLAT];             // [0..15]=EY, [16..271]=EYYt
  __shared__ float redA[NWAVE], redB[NWAVE];
  __shared__ unsigned int ep;
  if (tid == 0) ep = 0;

  // -------- per-particle state lives in registers --------
  float x[DLAT], y[DLAT], m[DLAT];
  ld16(x, x0 + (size_t)u*DLAT);
  ld16(y, y0 + (size_t)u*DLAT);
  ld16(m, m0 + (size_t)u*DLAT);
  float tl = 0.0f, tr = tau0[u], w = 1.0f / UPART;

  const float sigx = expf(lsx[0]);
  const float sigy = expf(lsy[0]);
  const float snDT = sigy * sqrtf(DT_F);

  // observation constants
  float c0 = 0.0f;
#pragma unroll
  for (int mm = 0; mm < MOBS_; ++mm)
    c0 += logf(fmaxf(expf(logR[mm]), 1e-10f));
  c0 += MOBS_ * 1.8378770664093453f;  // M*log(2*pi)
  const float rinv0 = 1.0f / fmaxf(expf(logR[ln]),      1e-10f);
  const float rinv1 = 1.0f / fmaxf(expf(logR[16 + ln]), 1e-10f);

  // B-matrix (W^T) tiles in WMMA 4x16 f32 layout, constant across steps:
  // lane holds (K = kc*4 + 2h [+1], N = nt*16 + ln); element = W[N][K]
  v2f bW[4][2];
#pragma unroll
  for (int kc = 0; kc < 4; ++kc)
#pragma unroll
    for (int nt = 0; nt < 2; ++nt) {
      int mrow = nt*16 + ln;
      int dcol = kc*4 + 2*h;
      v2f b; b.x = Wmat[mrow*DLAT + dcol]; b.y = Wmat[mrow*DLAT + dcol + 1];
      bW[kc][nt] = b;
    }
  __syncthreads();

  for (int k = 0; k < KSTEPS; ++k) {
    const float tk = k * DT_F;
    const size_t base = (size_t)k * UPART + u;

    if (k + 1 < KSTEPS) {  // prefetch next step's HBM streams (global_prefetch_b8)
      __builtin_prefetch(eps_x  + (base + UPART) * DLAT, 0, 0);
      __builtin_prefetch(eps_y  + (base + UPART) * DLAT, 0, 0);
      __builtin_prefetch(m_pool + (base + UPART) * DLAT, 0, 0);
    }

    // ---- renewal + SDE update (pure registers) ----
    const float tauk = tau_pool[base];
    const bool need = tk > tr;
    if (need) { tl = tr; tr = tl + tauk; }

    float mk[DLAT], exr[DLAT], eyr[DLAT];
    ld16(mk,  m_pool + base * DLAT);
    ld16(exr, eps_x  + base * DLAT);
    ld16(eyr, eps_y  + base * DLAT);

    const float denom = fmaxf(tr - tk, 1e-8f);
    const float bse   = fmaxf((tr - tk) * (tk - tl) / fmaxf(tr - tl, 1e-8f), 0.0f) * DT_F;
    const float nstd  = sigx * sqrtf(bse);
#pragma unroll
    for (int d = 0; d < DLAT; ++d) {
      if (need) m[d] = mk[d];
      x[d] += (m[d] - x[d]) / denom * DT_F + exr[d] * nstd;
      y[d] += x[d] * DT_F + eyr[d] * snDT;
    }

    // ---- observation log-weight via V_WMMA_F32_16X16X4_F32 ----
    // stage y (wave-local, DS ops are in-order per wave)
#pragma unroll
    for (int d = 0; d < DLAT; ++d) yv[wid][lane*DLAT + d] = y[d];

    const float zz[2] = { Z[k*MOBS_ + ln], Z[k*MOBS_ + 16 + ln] };
    const float ri[2] = { rinv0, rinv1 };

#pragma unroll
    for (int tile = 0; tile < 2; ++tile) {
      const int row = tile*16 + ln;           // A: M = particle row of this tile
      float acc[8];
#pragma unroll
      for (int r = 0; r < 8; ++r) acc[r] = 0.0f;
#pragma unroll
      for (int nt = 0; nt < 2; ++nt) {
        v8f c = {};
#pragma unroll
        for (int kc = 0; kc < 4; ++kc) {
          const int ad = row*DLAT + kc*4 + 2*h;   // (K = kc*4+2h, +1)
          v2f a; a.x = yv[wid][ad]; a.y = yv[wid][ad + 1];
          c = __builtin_amdgcn_wmma_f32_16x16x4_f32(
              false, a, false, bW[kc][nt], (short)0, c, false, false);
        }
#pragma unroll
        for (int r = 0; r < 8; ++r) { float df = zz[nt] - c[r]; acc[r] += df*df*ri[nt]; }
      }
      // reduce over N within each 16-lane half (lanes<16: M=r, lanes>=16: M=r+8)
#pragma unroll
      for (int r = 0; r < 8; ++r) {
        acc[r] += __shfl_xor(acc[r], 1, 32);
        acc[r] += __shfl_xor(acc[r], 2, 32);
        acc[r] += __shfl_xor(acc[r], 4, 32);
        acc[r] += __shfl_xor(acc[r], 8, 32);
      }
      if (lane == 0) {
#pragma unroll
        for (int r = 0; r < 8; ++r) lwl[wid][tile*16 + r] = acc[r];
      }
      if (lane == 16) {
#pragma unroll
        for (int r = 0; r < 8; ++r) lwl[wid][tile*16 + 8 + r] = acc[r];
      }
    }
    const float lw = -0.5f * (lwl[wid][lane] + c0);

    // ---- grid reduce: max(lw) ----
    float mx = lw;
#pragma unroll
    for (int msk = 1; msk < 32; msk <<= 1) mx = fmaxf(mx, __shfl_xor(mx, msk, 32));
    if (lane == 0) redA[wid] = mx;
    __syncthreads();
    if (tid == 0) {
      float m2 = redA[0];
      for (int i = 1; i < NWAVE; ++i) m2 = fmaxf(m2, redA[i]);
      gmax[blk] = m2;
    }
    grid_sync(bar, &ep);
    float M = gmax[0];
#pragma unroll
    for (int i = 1; i < NBLK; ++i) M = fmaxf(M, gmax[i]);

    // ---- weight update + grid reduce: S1, S2 ----
    const float wt = w * expf(lw - M);
    float a1 = wt, a2 = wt * wt;
#pragma unroll
    for (int msk = 1; msk < 32; msk <<= 1) {
      a1 += __shfl_xor(a1, msk, 32);
      a2 += __shfl_xor(a2, msk, 32);
    }
    if (lane == 0) { redA[wid] = a1; redB[wid] = a2; }
    __syncthreads();
    if (tid == 0) {
      float s1 = 0.0f, s2 = 0.0f;
      for (int i = 0; i < NWAVE; ++i) { s1 += redA[i]; s2 += redB[i]; }
      s1a[blk] = s1; s2a[blk] = s2;
    }
    grid_sync(bar, &ep);
    float S1 = 0.0f, S2 = 0.0f, bbase = 0.0f;
#pragma unroll
    for (int i = 0; i < NBLK; ++i) {
      float v = s1a[i];
      S1 += v;
      if (i < blk) bbase += v;
      S2 += s2a[i];
    }
    const float S1c = fmaxf(S1, 1e-12f);
    w = wt / S1c;
    const float ess = (S1 * S1) / fmaxf(S2, 1e-12f);

    // ---- conditional multinomial resample (uniform branch) ----
    if (ess < 0.5f * UPART) {
      // inclusive scan of wt: wave scan + block base + grid base
      float sc = wt;
#pragma unroll
      for (int off = 1; off < 32; off <<= 1) {
        float t = __shfl_up(sc, off, 32);
        if (lane >= off) sc += t;
      }
      if (lane == 31) redA[wid] = sc;
      __syncthreads();
      float wb = 0.0f;
      for (int i = 0; i < wid; ++i) wb += redA[i];
      cum[u] = (bbase + wb + sc) / S1c;
      st16(xs + (size_t)u*DLAT, x);
      st16(ys + (size_t)u*DLAT, y);
      st16(ms + (size_t)u*DLAT, m);
      tls[u] = tl; trs[u] = tr;
      grid_sync(bar, &ep);

      const float ru = resample_u[(size_t)k * UPART + u];
      int lo = 0, hi = UPART;           // searchsorted 'left'
      while (lo < hi) { int mid = (lo + hi) >> 1; if (cum[mid] < ru) lo = mid + 1; else hi = mid; }
      const int idx = min(lo, UPART - 1);
      ld16(x, xs + (size_t)idx*DLAT);
      ld16(y, ys + (size_t)idx*DLAT);
      ld16(m, ms + (size_t)idx*DLAT);
      tl = tls[idx]; tr = trs[idx];
      w = 1.0f / UPART;
    }

    // ---- posterior moments: EY = sum w*y ; EYYt = (w*y)^T . y via WMMA ----
    for (int i = tid; i < DLAT + DLAT*DLAT; i += NTHR) mom[i] = 0.0f;
#pragma unroll
    for (int d = 0; d < DLAT; ++d) {
      yv[wid][lane*DLAT + d] = y[d];
      yw[wid][lane*DLAT + d] = w * y[d];
    }
    __syncthreads();
    {
      v8f c = {};
#pragma unroll
      for (int c0i = 0; c0i < 32; c0i += 4) {     // K = particle index within wave
        const int ad = (c0i + 2*h) * DLAT + ln;
        v2f a; a.x = yw[wid][ad]; a.y = yw[wid][ad + DLAT];   // A[d, j] = w_j*y_j[d]
        v2f b; b.x = yv[wid][ad]; b.y = yv[wid][ad + DLAT];   // B[j, e] = y_j[e]
        c = __builtin_amdgcn_wmma_f32_16x16x4_f32(
            false, a, false, b, (short)0, c, false, false);
      }
#pragma unroll
      for (int r = 0; r < 8; ++r)
        atomicAdd(&mom[DLAT + (r + 8*h)*DLAT + ln], c[r]);    // ds_add_f32
    }
#pragma unroll
    for (int d = 0; d < DLAT; ++d) {
      float v = w * y[d];
#pragma unroll
      for (int msk = 1; msk < 32; msk <<= 1) v += __shfl_xor(v, msk, 32);
      if (lane == 0) atomicAdd(&mom[d], v);
    }
    __syncthreads();
    if (tid < DLAT)
      unsafeAtomicAdd(&out[(size_t)k*DLAT + tid], mom[tid]);
    if (tid < DLAT*DLAT)
      unsafeAtomicAdd(&out[KSTEPS*DLAT + (size_t)k*DLAT*DLAT + tid], mom[DLAT + tid]);
    __syncthreads();   // protect mom reuse next step
  }
}

extern "C" void kernel_launch(void* const* d_in, const int* in_sizes, int n_in,
                              void* d_out, int out_size, void* d_ws, size_t ws_size,
                              hipStream_t stream) {
  (void)in_sizes; (void)n_in; (void)ws_size;
  // Deterministic per-launch init (graph-capturable): barrier header + output.
  hipMemsetAsync(d_ws, 0, 256, stream);
  hipMemsetAsync(d_out, 0, (size_t)out_size * sizeof(float), stream);

  const float* Z          = (const float*)d_in[0];
  const float* x0         = (const float*)d_in[1];
  const float* y0         = (const float*)d_in[2];
  const float* tau0       = (const float*)d_in[3];
  const float* m0         = (const float*)d_in[4];
  const float* tau_pool   = (const float*)d_in[5];
  const float* m_pool     = (const float*)d_in[6];
  const float* eps_x      = (const float*)d_in[7];
  const float* eps_y      = (const float*)d_in[8];
  const float* resample_u = (const float*)d_in[9];
  const float* Wmat       = (const float*)d_in[10];
  const float* logR       = (const float*)d_in[11];
  const float* lsx        = (const float*)d_in[12];
  const float* lsy        = (const float*)d_in[13];

  unsigned int* bar = (unsigned int*)d_ws;
  float* wsf = (float*)((char*)d_ws + 256);

  pf_persistent<<<NBLK, NTHR, 0, stream>>>(
      Z, x0, y0, tau0, m0, tau_pool, m_pool, eps_x, eps_y, resample_u,
      Wmat, logR, lsx, lsy, (float*)d_out, bar, wsf);
}